// GaussianModel_74457553044377
// MI455X (gfx1250) — compile-verified
//
#include <hip/hip_runtime.h>

typedef float float4v __attribute__((ext_vector_type(4)));

// Each thread handles 4 gaussians:
//   loads : 4 x B128 (rotation) + 3 x B128 (scale)   -- all NT, coalesced
//   stores: 9 x B128 (covariance)                    -- all NT, coalesced
__global__ __launch_bounds__(256) void quat_cov4_kernel(
    const float* __restrict__ rot,   // (N,4)
    const float* __restrict__ scl,   // (N,3)
    float* __restrict__ out,         // (N,3,3)
    int nquad)                       // N/4
{
    int t = blockIdx.x * blockDim.x + threadIdx.x;
    if (t >= nquad) return;
    size_t tb = (size_t)t;

    // ---- load 4 quaternions (16 floats) ----
    const float4v* rp = (const float4v*)(rot + 16 * tb);
    float4v q[4];
#pragma unroll
    for (int i = 0; i < 4; ++i) q[i] = __builtin_nontemporal_load(rp + i);

    // ---- load 4 scales (12 floats) ----
    const float4v* sp = (const float4v*)(scl + 12 * tb);
    float s[12];
#pragma unroll
    for (int i = 0; i < 3; ++i) {
        float4v v = __builtin_nontemporal_load(sp + i);
#pragma unroll
        for (int j = 0; j < 4; ++j) s[4 * i + j] = v[j];
    }

    // ---- compute 4 covariances (36 floats) ----
    float o[36];
#pragma unroll
    for (int g = 0; g < 4; ++g) {
        float w = q[g][0], x = q[g][1], y = q[g][2], z = q[g][3];
        // fold normalization: 2*q̂i*q̂j == sc*qi*qj with sc = 2/|q|^2.
        // sc = rcp(0.5*|q|^2) via hardware v_rcp_f32 (~1 ulp), avoiding the
        // ~8-instruction IEEE division sequence per gaussian.
        float n  = w * w + x * x + y * y + z * z;
        float sc = __builtin_amdgcn_rcpf(0.5f * n);

        float xx = x * x, yy = y * y, zz = z * z;
        float xy = x * y, xz = x * z, yz = y * z;
        float wx = w * x, wy = w * y, wz = w * z;

        float R[3][3];
        R[0][0] = 1.0f - sc * (yy + zz);
        R[0][1] = sc * (xy - wz);
        R[0][2] = sc * (xz + wy);
        R[1][0] = sc * (xy + wz);
        R[1][1] = 1.0f - sc * (xx + zz);
        R[1][2] = sc * (yz - wx);
        R[2][0] = sc * (xz - wy);
        R[2][1] = sc * (yz + wx);
        R[2][2] = 1.0f - sc * (xx + yy);

        float m[3][3];
#pragma unroll
        for (int i = 0; i < 3; ++i)
#pragma unroll
            for (int j = 0; j < 3; ++j) m[i][j] = R[i][j] * s[3 * g + j];

        // cov = m * m^T (symmetric: 6 unique dot products)
#pragma unroll
        for (int i = 0; i < 3; ++i)
#pragma unroll
            for (int k = i; k < 3; ++k) {
                float c = m[i][0] * m[k][0] + m[i][1] * m[k][1] + m[i][2] * m[k][2];
                o[9 * g + 3 * i + k] = c;
                o[9 * g + 3 * k + i] = c;
            }
    }

    // ---- store 36 floats as 9 x B128 NT ----
    float4v* op = (float4v*)(out + 36 * tb);
#pragma unroll
    for (int i = 0; i < 9; ++i) {
        float4v v = {o[4 * i + 0], o[4 * i + 1], o[4 * i + 2], o[4 * i + 3]};
        __builtin_nontemporal_store(v, op + i);
    }
}

// Scalar tail for N % 4 leftover gaussians (not hit for N = 4,000,000).
__global__ void quat_cov_tail_kernel(const float* __restrict__ rot,
                                     const float* __restrict__ scl,
                                     float* __restrict__ out,
                                     int base, int n)
{
    int i = base + blockIdx.x * blockDim.x + threadIdx.x;
    if (i >= n) return;
    size_t ib = (size_t)i;

    float w = rot[4 * ib + 0], x = rot[4 * ib + 1];
    float y = rot[4 * ib + 2], z = rot[4 * ib + 3];
    float s0 = scl[3 * ib + 0], s1 = scl[3 * ib + 1], s2 = scl[3 * ib + 2];

    float n2 = w * w + x * x + y * y + z * z;
    float sc = __builtin_amdgcn_rcpf(0.5f * n2);
    float xx = x * x, yy = y * y, zz = z * z;
    float xy = x * y, xz = x * z, yz = y * z;
    float wx = w * x, wy = w * y, wz = w * z;

    float m[3][3];
    m[0][0] = (1.0f - sc * (yy + zz)) * s0;
    m[0][1] = (sc * (xy - wz)) * s1;
    m[0][2] = (sc * (xz + wy)) * s2;
    m[1][0] = (sc * (xy + wz)) * s0;
    m[1][1] = (1.0f - sc * (xx + zz)) * s1;
    m[1][2] = (sc * (yz - wx)) * s2;
    m[2][0] = (sc * (xz - wy)) * s0;
    m[2][1] = (sc * (yz + wx)) * s1;
    m[2][2] = (1.0f - sc * (xx + yy)) * s2;

    for (int r = 0; r < 3; ++r)
        for (int k = 0; k < 3; ++k)
            out[9 * ib + 3 * r + k] =
                m[r][0] * m[k][0] + m[r][1] * m[k][1] + m[r][2] * m[k][2];
}

extern "C" void kernel_launch(void* const* d_in, const int* in_sizes, int n_in,
                              void* d_out, int out_size, void* d_ws, size_t ws_size,
                              hipStream_t stream) {
    (void)n_in; (void)d_ws; (void)ws_size; (void)out_size;

    const float* rot = (const float*)d_in[0];   // (N,4) f32
    const float* scl = (const float*)d_in[1];   // (N,3) f32
    float* out = (float*)d_out;                 // (N,3,3) f32

    int N = in_sizes[0] / 4;                    // rotation flat count = 4N
    int nquad = N / 4;
    int rem = N - 4 * nquad;

    if (nquad > 0) {
        int blocks = (nquad + 255) / 256;
        quat_cov4_kernel<<<blocks, 256, 0, stream>>>(rot, scl, out, nquad);
    }
    if (rem > 0) {
        quat_cov_tail_kernel<<<1, 64, 0, stream>>>(rot, scl, out, 4 * nquad, N);
    }
}